// SimpleInterestClock_74448963109134
// MI455X (gfx1250) — compile-verified
//
#include <hip/hip_runtime.h>

typedef __attribute__((ext_vector_type(2))) float v2f;
typedef __attribute__((ext_vector_type(8))) float v8f;

#define B_  4096
#define L_  200
#define D_  256
#define LT_ 208   // L padded to tile multiple

__device__ inline float blk_max(float v, volatile float* red) {
    const int t = threadIdx.x;
    red[t] = v; __syncthreads();
    #pragma unroll
    for (int s = 128; s > 0; s >>= 1) {
        if (t < s) red[t] = fmaxf(red[t], red[t + s]);
        __syncthreads();
    }
    const float r = red[0];
    __syncthreads();
    return r;
}

__device__ inline float blk_sum(float v, volatile float* red) {
    const int t = threadIdx.x;
    red[t] = v; __syncthreads();
    #pragma unroll
    for (int s = 128; s > 0; s >>= 1) {
        if (t < s) red[t] = red[t] + red[t + s];
        __syncthreads();
    }
    const float r = red[0];
    __syncthreads();
    return r;
}

__global__ __launch_bounds__(256) void sic_kernel(
    const int* __restrict__ items,          // [B,L]
    const int* __restrict__ dts,            // [B,L]
    const unsigned char* __restrict__ mask, // [B,L] (bool)
    const int* __restrict__ pos_items,      // [B]
    const int* __restrict__ neg_items,      // [B]
    const float* __restrict__ item_emb,     // [NUM_ITEMS, D]
    const float* __restrict__ dt_gate,      // [NUM_DT]
    const float* __restrict__ raw_tau,      // [1]
    float* __restrict__ out)                // pos[B] | neg[B] | attn[B,L]
{
    // s_q: [0..255] = q_pos, [256..511] = q_neg, [512..767] = zeros
    __shared__ float s_q[3 * D_];
    __shared__ float s_simpos[LT_];
    __shared__ float s_simneg[LT_];
    __shared__ float s_red[256];

    const int b    = blockIdx.x;
    const int tid  = threadIdx.x;
    const int lane = tid & 31;
    const int wave = tid >> 5;

    // Stage candidate embeddings + zero pad region (D_ == blockDim.x == 256)
    {
        const long qp = (long)pos_items[b] * D_;
        const long qn = (long)neg_items[b] * D_;
        s_q[tid]          = item_emb[qp + tid];
        s_q[D_ + tid]     = item_emb[qn + tid];
        s_q[2 * D_ + tid] = 0.f;
    }
    __syncthreads();

    const int hl = lane >> 4;   // K-half selector (A/B: K={0,1} vs {2,3})
    const int m  = lane & 15;   // A row within tile / B column index

    // Branchless per-lane B base: col 0 -> q_pos, col 1 -> q_neg, cols 2..15 -> zeros
    const float* qbase = s_q + ((m < 2) ? m : 2) * D_ + 2 * hl;

    // sim via fp32 WMMA: each wave owns 16-row tiles; N=0 col = q_pos, N=1 col = q_neg
    for (int t = wave; t * 16 < L_; t += 8) {
        const int l0 = t * 16;
        const int l  = l0 + m;
        const int row = (l < L_) ? items[(long)b * L_ + l] : 0;
        const float* arow = item_emb + (long)row * D_ + 2 * hl;

        // prefetch the row this lane will need next tile round (global_prefetch_b8)
        {
            const int lp = l0 + 128 + m;   // tile t+8
            if (lp < L_) {
                const int prow = items[(long)b * L_ + lp];
                __builtin_prefetch(item_emb + (long)prow * D_, 0, 3);
            }
        }

        v8f acc = {0.f, 0.f, 0.f, 0.f, 0.f, 0.f, 0.f, 0.f};
        #pragma unroll 8
        for (int kk = 0; kk < D_; kk += 4) {
            const v2f a  = *reinterpret_cast<const v2f*>(arow + kk);   // global_load_b64
            const v2f bb = *reinterpret_cast<const v2f*>(qbase + kk);  // ds_load_b64
            acc = __builtin_amdgcn_wmma_f32_16x16x4_f32(
                false, a, false, bb, (short)0, acc, false, false);
        }

        // C layout: lane N=m; VGPR r -> row l0+r (lanes 0-15) or l0+8+r (lanes 16-31)
        if (m == 0) {
            #pragma unroll
            for (int r = 0; r < 8; ++r) s_simpos[l0 + 8 * hl + r] = acc[r];
        } else if (m == 1) {
            #pragma unroll
            for (int r = 0; r < 8; ++r) s_simneg[l0 + 8 * hl + r] = acc[r];
        }
    }
    __syncthreads();

    // tau = softplus(raw_tau) + 1e-6 (numerically stable)
    const float rt  = raw_tau[0];
    const float tau = fmaxf(rt, 0.f) + log1pf(expf(-fabsf(rt))) + 1e-6f;
    const float inv_tau = 1.f / tau;

    // Per-position logits (one l per thread)
    float vp = -INFINITY, vn = -INFINITY;   // logits
    float sp = 0.f, sn = 0.f;               // raw sims (for score = sum attn*sim)
    if (tid < L_) {
        const long idx = (long)b * L_ + tid;
        if (mask[idx]) {
            const float g = dt_gate[dts[idx]];
            sp = s_simpos[tid];
            sn = s_simneg[tid];
            vp = sp * g * inv_tau;
            vn = sn * g * inv_tau;
        }
    }

    const float mp = blk_max(vp, s_red);
    const float mn = blk_max(vn, s_red);
    const float ep = expf(vp - mp);   // 0 for masked/padded when mp finite
    const float en = expf(vn - mn);
    const float Zp = blk_sum(ep, s_red);
    const float Zn = blk_sum(en, s_red);
    const float ap = ep / Zp;
    const float an = en / Zn;

    if (tid < L_) out[2 * B_ + (long)b * L_ + tid] = ap;

    // pos_score = sum_l attn_pos[l] * sim_pos[l]  (u never materialized)
    const float ps = blk_sum((tid < L_) ? ap * sp : 0.f, s_red);
    const float ns = blk_sum((tid < L_) ? an * sn : 0.f, s_red);
    if (tid == 0) {
        out[b]      = ps;
        out[B_ + b] = ns;
    }
}

extern "C" void kernel_launch(void* const* d_in, const int* in_sizes, int n_in,
                              void* d_out, int out_size, void* d_ws, size_t ws_size,
                              hipStream_t stream) {
    const int*           items = (const int*)d_in[0];
    const int*           dts   = (const int*)d_in[1];
    const unsigned char* msk   = (const unsigned char*)d_in[2];
    const int*           pos   = (const int*)d_in[3];
    const int*           neg   = (const int*)d_in[4];
    const float*         emb   = (const float*)d_in[5];
    const float*         gate  = (const float*)d_in[6];
    const float*         rtau  = (const float*)d_in[7];
    float*               out   = (float*)d_out;

    sic_kernel<<<B_, 256, 0, stream>>>(items, dts, msk, pos, neg, emb, gate, rtau, out);
}